// CapsLayer_33964601377223
// MI455X (gfx1250) — compile-verified
//
#include <hip/hip_runtime.h>

// CapsNet dynamic-routing layer for MI455X (gfx1250, wave32, WMMA).
//
// Sizes: x[64][2048][8] f32, W[2048][32][8][16] f32, out v[64][32][16] f32.
//
// Strategy: never materialize u_hat (268MB > L2). Instead:
//   pass A: s0 = (1/32) * (X2[64x16384] @ W2[16384x512])   (pure WMMA GEMM)
//   v0 = squash(s0)
//   pass B: per (b,i): u = x2[b, i*8..]·W2 block (WMMA), logits = u.v0,
//           c = softmax_j, s1 += c ⊙ u   (atomics over i-blocks)
//   v1 = squash(s1); vsum = v0+v1   (logits are linear in v => b never stored)
//   pass C: same as B with vsum; v2 = squash(s2) -> d_out
// HBM traffic ~ W(33.5MB)+x(4MB); x/W re-reads served by 192MB L2.

typedef float v2f __attribute__((ext_vector_type(2)));
typedef float v8f __attribute__((ext_vector_type(8)));

#define B_SZ    64
#define IN_CAPS 2048
#define IN_DIM  8
#define NC      32
#define OD      16
#define KDIM    (IN_CAPS * IN_DIM)   // 16384
#define NDIM    (NC * OD)            // 512
#define WROW    (NC * IN_DIM * OD)   // 4096 floats per in-capsule i

__device__ __forceinline__ v8f v8f_zero() {
    v8f z;
    #pragma unroll
    for (int i = 0; i < 8; ++i) z[i] = 0.f;
    return z;
}

// D = A(16x4 f32) * B(4x16 f32) + C(16x16 f32), wave32.
// A/B: 2 VGPRs each: lanes 0-15 hold K=0,1 ; lanes 16-31 hold K=2,3.
// C/D: 8 VGPRs: vgpr r, lanes 0-15 -> (M=r, N=lane); lanes 16-31 -> (M=r+8, N=lane-16).
__device__ __forceinline__ v8f wmma_f32(v2f a, v2f b, v8f c) {
    return __builtin_amdgcn_wmma_f32_16x16x4_f32(
        /*neg_a=*/false, a, /*neg_b=*/false, b,
        /*c_mod=*/(short)0, c, /*reuse_a=*/false, /*reuse_b=*/false);
}

// ---------------------------------------------------------------------------
// Pass A: s += X2 @ W2.  X2[b, k=(i*8+d)] = x[b,i,d]; W2[k, n=(j*16+e)] = W[i,j,d,e].
// grid: (32 j-tiles, 8 k-blocks), block 256 (8 waves), each wave K-slice = 256.
// Inner loop: kk steps by 8 -> W advances by a constant 4096 floats, x by 8;
// everything else is immediate offsets. 8 WMMA + 1 prefetch per iteration.
// ---------------------------------------------------------------------------
__global__ __launch_bounds__(256) void gemm_s0_kernel(const float* __restrict__ x,
                                                      const float* __restrict__ W,
                                                      float* __restrict__ s) {
    const int j    = blockIdx.x;            // capsule = N-tile (n = j*16 + e)
    const int lane = threadIdx.x & 31;
    const int wave = threadIdx.x >> 5;      // 0..7
    const int e    = lane & 15;
    const int hi   = (lane >> 4) & 1;       // K-half select within WMMA operand
    const int mrow = lane & 15;

    v8f acc[4];
    #pragma unroll
    for (int mt = 0; mt < 4; ++mt) acc[mt] = v8f_zero();

    const int k0  = blockIdx.y * (KDIM / 8) + wave * (KDIM / 64); // 256 K values
    const int kk0 = k0 + 2 * hi;            // this lane-half's first K index
    // kk0 is a multiple of 256 plus 2*hi, so (kk0>>3) = i0, (kk0&7) = 2*hi.
    const float* wp = W + ((((size_t)(kk0 >> 3) * NC + j) * IN_DIM + (kk0 & 7)) * OD + e);
    const float* xp = x + ((size_t)mrow * KDIM + kk0);

    #pragma unroll 1
    for (int it = 0; it < (KDIM / 64) / 8; ++it) {   // 32 iterations
        // speculative prefetch of the W stream 4 iterations (16KB/row) ahead
        __builtin_prefetch(wp + (size_t)4 * WROW, 0, 1);
        v2f b0; b0.x = wp[0];      b0.y = wp[OD];        // d, d+1
        v2f b1; b1.x = wp[4 * OD]; b1.y = wp[5 * OD];    // d+4, d+5
        #pragma unroll
        for (int mt = 0; mt < 4; ++mt) {
            const float* xq = xp + (size_t)mt * 16 * KDIM;  // 24-bit imm offsets
            v2f a0; a0.x = xq[0]; a0.y = xq[1];
            v2f a1; a1.x = xq[4]; a1.y = xq[5];
            acc[mt] = wmma_f32(a0, b0, acc[mt]);
            acc[mt] = wmma_f32(a1, b1, acc[mt]);
        }
        wp += WROW;   // +4096 floats: next pair of k-steps (i += 1)
        xp += 8;
    }

    // Reduce the 8 waves' partial tiles in LDS, then atomically add to s.
    __shared__ float red[8][4][256];
    #pragma unroll
    for (int mt = 0; mt < 4; ++mt)
        #pragma unroll
        for (int r = 0; r < 8; ++r)
            red[wave][mt][r * 32 + lane] = acc[mt][r];
    __syncthreads();

    const int t = threadIdx.x;
    const int r = t >> 5, l = t & 31;
    const int m = r + ((l & 16) ? 8 : 0);
    const int n = l & 15;
    #pragma unroll
    for (int mt = 0; mt < 4; ++mt) {
        float sum = 0.f;
        #pragma unroll
        for (int w = 0; w < 8; ++w) sum += red[w][mt][t];
        atomicAdd(&s[(size_t)(mt * 16 + m) * NDIM + j * 16 + n], sum);
    }
}

// ---------------------------------------------------------------------------
// squash: v = squash(s * scale); if vprev != null: vout = vprev + v.
// One thread per (b,j) group of 16 elems. 2048 groups.
// ---------------------------------------------------------------------------
__global__ void squash_kernel(const float* __restrict__ s, float scale,
                              const float* vprev, float* vout) {
    const int g = blockIdx.x * blockDim.x + threadIdx.x;
    if (g >= B_SZ * NC) return;
    const float* sp = s + (size_t)g * OD;
    float vals[OD];
    float s2 = 0.f;
    #pragma unroll
    for (int e = 0; e < OD; ++e) {
        float v = sp[e] * scale;
        vals[e] = v;
        s2 += v * v;
    }
    const float f = sqrtf(s2) / (1.f + s2);   // == s2/(1+s2)/sqrt(s2)
    float* op = vout + (size_t)g * OD;
    if (vprev) {
        const float* pp = vprev + (size_t)g * OD;
        #pragma unroll
        for (int e = 0; e < OD; ++e) op[e] = pp[e] + f * vals[e];
    } else {
        #pragma unroll
        for (int e = 0; e < OD; ++e) op[e] = f * vals[e];
    }
}

// ---------------------------------------------------------------------------
// Routing pass: for each i: u = WMMA(x_i, W_i); logit[b,j] = u.vbuf;
// c = softmax_j(logit); s += c ⊙ u  (accumulated in regs, atomicAdd at end).
// grid: (i-blocks, 4 m-tiles), block 256 (8 waves x 4 capsules each).
// ---------------------------------------------------------------------------
__global__ __launch_bounds__(256) void routing_pass_kernel(
    const float* __restrict__ x, const float* __restrict__ W,
    const float* __restrict__ vbuf, float* __restrict__ s, int ichunk) {
    __shared__ float Llds[16][NC];   // logits  [m][j]
    __shared__ float Clds[16][NC];   // softmax [m][j]

    const int mtile = blockIdx.y;
    const int lane  = threadIdx.x & 31;
    const int wave  = threadIdx.x >> 5;
    const int jbase = wave * 4;
    const int e     = lane & 15;
    const int hi    = (lane >> 4) & 1;
    const int mrow  = lane & 15;
    const int mC    = (lane & 16) ? 8 : 0;   // C-layout row offset for this lane

    // v tile in WMMA C layout: vt[jj][r] = vbuf[m(r,lane)][jbase+jj][e]
    float vt[4][8];
    v8f acc[4];
    #pragma unroll
    for (int jj = 0; jj < 4; ++jj) {
        acc[jj] = v8f_zero();
        const int j = jbase + jj;
        #pragma unroll
        for (int r = 0; r < 8; ++r)
            vt[jj][r] = vbuf[(size_t)(mtile * 16 + r + mC) * NDIM + j * 16 + e];
    }

    const int i0 = blockIdx.x * ichunk;
    // Strength-reduced streams: per i, x advances by 8 floats, W by 4096.
    const float* xp = x + ((size_t)(mtile * 16 + mrow) * KDIM + (size_t)i0 * 8 + 2 * hi);
    const float* wp = W + (((size_t)i0 * NC + jbase) * IN_DIM + 2 * hi) * OD + e;

    #pragma unroll 1
    for (int ii = 0; ii < ichunk; ++ii) {
        __builtin_prefetch(wp + (size_t)2 * WROW, 0, 1);
        // A operand for both k-steps of the K=8 contraction over d
        v2f a0; a0.x = xp[0]; a0.y = xp[1];
        v2f a1; a1.x = xp[4]; a1.y = xp[5];

        v8f u[4];
        #pragma unroll
        for (int jj = 0; jj < 4; ++jj) {
            const float* wq = wp + jj * IN_DIM * OD;   // +128 floats per capsule
            v2f b0; b0.x = wq[0];      b0.y = wq[OD];
            v2f b1; b1.x = wq[4 * OD]; b1.y = wq[5 * OD];
            v8f c = v8f_zero();
            c = wmma_f32(a0, b0, c);
            c = wmma_f32(a1, b1, c);
            u[jj] = c;
        }

        // logits: reduce u*v over e (16-lane halves reduce independently)
        #pragma unroll
        for (int jj = 0; jj < 4; ++jj) {
            const int j = jbase + jj;
            #pragma unroll
            for (int r = 0; r < 8; ++r) {
                float p = u[jj][r] * vt[jj][r];
                #pragma unroll
                for (int off = 1; off < 16; off <<= 1)
                    p += __shfl_xor(p, off, 16);
                if (e == 0) Llds[r + mC][j] = p;   // lanes 0 and 16 write
            }
        }
        __syncthreads();

        // softmax over the 32 capsules, one thread per batch row m
        if (threadIdx.x < 16) {
            const int m = threadIdx.x;
            float mx = Llds[m][0];
            #pragma unroll
            for (int j = 1; j < NC; ++j) mx = fmaxf(mx, Llds[m][j]);
            float ex[NC];
            float den = 0.f;
            #pragma unroll
            for (int j = 0; j < NC; ++j) { ex[j] = __expf(Llds[m][j] - mx); den += ex[j]; }
            const float inv = 1.f / den;
            #pragma unroll
            for (int j = 0; j < NC; ++j) Clds[m][j] = ex[j] * inv;
        }
        __syncthreads();

        // s += c ⊙ u
        #pragma unroll
        for (int jj = 0; jj < 4; ++jj) {
            const int j = jbase + jj;
            #pragma unroll
            for (int r = 0; r < 8; ++r)
                acc[jj][r] += Clds[r + mC][j] * u[jj][r];
        }
        __syncthreads();   // protect Clds/Llds before next iteration rewrites

        xp += IN_DIM;      // next in-capsule
        wp += WROW;
    }

    #pragma unroll
    for (int jj = 0; jj < 4; ++jj) {
        const int j = jbase + jj;
        #pragma unroll
        for (int r = 0; r < 8; ++r)
            atomicAdd(&s[(size_t)(mtile * 16 + r + mC) * NDIM + j * 16 + e],
                      acc[jj][r]);
    }
}

// ---------------------------------------------------------------------------
extern "C" void kernel_launch(void* const* d_in, const int* in_sizes, int n_in,
                              void* d_out, int out_size, void* d_ws, size_t ws_size,
                              hipStream_t stream) {
    (void)in_sizes; (void)n_in; (void)out_size; (void)ws_size;
    const float* x = (const float*)d_in[0];   // [64][2048][8]
    const float* W = (const float*)d_in[1];   // [2048][32][8][16]
    float* out  = (float*)d_out;              // [64][32][16]
    float* s    = (float*)d_ws;               // [64][512] accumulator
    float* vbuf = s + (size_t)B_SZ * NDIM;    // [64][512] v / (v0+v1)
    const size_t sbytes = (size_t)B_SZ * NDIM * sizeof(float);

    // Pass A: s0 = X2 @ W2 (uniform routing weights fold into 1/32 scale)
    hipMemsetAsync(s, 0, sbytes, stream);
    gemm_s0_kernel<<<dim3(32, 8), 256, 0, stream>>>(x, W, s);
    squash_kernel<<<dim3(8), 256, 0, stream>>>(s, 1.f / 32.f, nullptr, vbuf); // v0

    // Pass B: logits = u.v0
    hipMemsetAsync(s, 0, sbytes, stream);
    routing_pass_kernel<<<dim3(64, 4), 256, 0, stream>>>(x, W, vbuf, s, 32);
    squash_kernel<<<dim3(8), 256, 0, stream>>>(s, 1.f, vbuf, vbuf);  // vbuf = v0+v1

    // Pass C: logits = u.(v0+v1)
    hipMemsetAsync(s, 0, sbytes, stream);
    routing_pass_kernel<<<dim3(64, 4), 256, 0, stream>>>(x, W, vbuf, s, 32);
    squash_kernel<<<dim3(8), 256, 0, stream>>>(s, 1.f, nullptr, out); // v2
}